// CrossModalAttention_19112604467909
// MI455X (gfx1250) — compile-verified
//
#include <hip/hip_runtime.h>
#include <math.h>

// ---------------- problem constants ----------------
#define MOD   4            // modalities
#define CCH   128          // channels
#define NHD   4            // heads
#define HD    32           // head dim
#define NTOK  1728         // tokens (12^3)
#define MN    (MOD*NTOK)   // 6912 concatenated K/V tokens
#define OUTC  128
#define NT16  (NTOK/16)    // 108 token tiles
#define NKT   (MN/16)      // 432 key tiles
#define BNEPS 1e-5f

typedef __attribute__((ext_vector_type(2))) float v2f;
typedef __attribute__((ext_vector_type(4))) float v4f;
typedef __attribute__((ext_vector_type(8))) float v8f;

// fp32 WMMA: D(16x16) = A(16x4) * B(4x16) + C   -- exact fp32 path on gfx1250
__device__ __forceinline__ v8f wmma4(v2f a, v2f b, v8f c) {
  return __builtin_amdgcn_wmma_f32_16x16x4_f32(false, a, false, b, (short)0, c,
                                               false, false);
}

// ---------------------------------------------------------------------------
// Kernel 0: zero the fused accumulator (ws is poisoned by the harness)
// ---------------------------------------------------------------------------
__global__ void cma_zero(float* __restrict__ p, int n) {
  int i = blockIdx.x * blockDim.x + threadIdx.x;
  if (i < n) p[i] = 0.0f;
}

// ---------------------------------------------------------------------------
// Kernel 1: QKV projections.  out[o,n] = sum_c W[m,o,c] * feat[m,c,n]
// One wave per 16x16 output tile; K-loop = 32 x v_wmma_f32_16x16x4_f32.
// Stored token-major [tok][HD] for contiguous float2 operand loads later:
//   Q: [(m*NH+h)*N + n][d]       K/V: [h*MN + m*N + n][d]
// ---------------------------------------------------------------------------
__global__ __launch_bounds__(256) void cma_proj(
    const float* __restrict__ f0, const float* __restrict__ f1,
    const float* __restrict__ f2, const float* __restrict__ f3,
    const float* __restrict__ Wq, const float* __restrict__ Wk,
    const float* __restrict__ Wv,
    float* __restrict__ Qt, float* __restrict__ Kt, float* __restrict__ Vt) {
  int wave = (blockIdx.x * blockDim.x + threadIdx.x) >> 5;
  int lane = threadIdx.x & 31;
  int nt = wave % NT16;      int r = wave / NT16;
  int ot = r % (CCH / 16);   r /= (CCH / 16);
  int m  = r % MOD;          int proj = r / MOD;   // 0=Q 1=K 2=V

  const float* X = (m == 0) ? f0 : (m == 1) ? f1 : (m == 2) ? f2 : f3;
  const float* W = ((proj == 0) ? Wq : (proj == 1) ? Wk : Wv) + m * CCH * CCH;

  int hl = lane >> 4, ll = lane & 15;
  int n = nt * 16 + ll;          // output column (token)
  int orow = ot * 16 + ll;       // A-matrix row (out channel)

  v8f acc = {0.f, 0.f, 0.f, 0.f, 0.f, 0.f, 0.f, 0.f};
  for (int c0 = 0; c0 < CCH; c0 += 4) {
    int c = c0 + 2 * hl;
    v2f a = *(const v2f*)(W + orow * CCH + c);      // W[o][c], c even -> b64
    v2f b;
    b.x = X[c * NTOK + n];                          // X[c][n]
    b.y = X[(c + 1) * NTOK + n];
    acc = wmma4(a, b, acc);
  }

  // C layout: vgpr v <-> row (v + 8*hl); column n per lane.
  int h  = ot >> 1;                     // head of this 16-row slab
  int d0 = (ot & 1) * 16 + 8 * hl;      // 8 consecutive d values
  float* dst;
  long tok;
  if (proj == 0) { dst = Qt; tok = (long)(m * NHD + h) * NTOK + n; }
  else           { dst = (proj == 1) ? Kt : Vt;
                   tok = (long)h * MN + (long)m * NTOK + n; }
  float* p = dst + tok * HD + d0;
  *(v4f*)p       = *(const v4f*)&acc;
  *(v4f*)(p + 4) = *((const v4f*)&acc + 1);
}

// ---------------------------------------------------------------------------
// Kernel 2: flash attention. Block = 4 waves = 4 consecutive q-tiles of one
// (modality i, head h). K/V tiles are staged ONCE per block into LDS with
// gfx1250 async global->LDS DMA (double-buffered, ASYNCcnt-tracked), so the
// hot loop reads operands via ds_load instead of 4x-redundant global loads.
//   S^T = K * Q^T  (rows = keys, cols = queries -> per-lane softmax state)
//   online softmax, then O^T += V^T * P^T
// Modality outputs summed into fused[C][N] via global_atomic_add_f32.
// ---------------------------------------------------------------------------
__global__ __launch_bounds__(128) void cma_attn(
    const float* __restrict__ Qt, const float* __restrict__ Kt,
    const float* __restrict__ Vt, float* __restrict__ fused) {
  __shared__ float bufK[2][16][HD];   // [buf][kk][d]  2 KB each
  __shared__ float bufV[2][16][HD];

  int bq = blockIdx.x % (NT16 / 4);   int r = blockIdx.x / (NT16 / 4);
  int h  = r % NHD;                   int i = r / NHD;

  int tid  = threadIdx.x;
  int wave = tid >> 5;
  int lane = tid & 31;
  int hl = lane >> 4, ll = lane & 15;
  int qt = bq * 4 + wave;
  int q  = qt * 16 + ll;

  const float* Kb = Kt + (long)h * MN * HD;
  const float* Vb = Vt + (long)h * MN * HD;

  // async stage helper indices: 128 threads x 16B cover one 16x32 f32 tile
  int skk = tid >> 3;            // 0..15
  int sd  = (tid & 7) * 4;       // 0,4,...,28

  // Preload Q^T as 8 B-operand chunks (d = 4*dc + 2*hl + {0,1})
  v2f qB[8];
  const float* Qbase = Qt + ((long)(i * NHD + h) * NTOK + q) * HD + 2 * hl;
  #pragma unroll
  for (int dc = 0; dc < 8; ++dc) qB[dc] = *(const v2f*)(Qbase + dc * 4);

  // prologue: async-stage key tile 0 into buffer 0
  {
    const float* gK = Kb + (long)skk * HD + sd;
    const float* gV = Vb + (long)skk * HD + sd;
    unsigned lK = (unsigned)(uintptr_t)&bufK[0][skk][sd];
    unsigned lV = (unsigned)(uintptr_t)&bufV[0][skk][sd];
    asm volatile("global_load_async_to_lds_b128 %0, %1, off"
                 :: "v"(lK), "v"(gK) : "memory");
    asm volatile("global_load_async_to_lds_b128 %0, %1, off"
                 :: "v"(lV), "v"(gV) : "memory");
  }

  const float invscale = 0.17677669529663687f;  // 1/sqrt(HD)
  float mrun = -INFINITY, lrun = 0.f;
  v8f oLo = {0.f,0.f,0.f,0.f,0.f,0.f,0.f,0.f};
  v8f oHi = {0.f,0.f,0.f,0.f,0.f,0.f,0.f,0.f};

  for (int kt = 0; kt < NKT; ++kt) {
    int cur = kt & 1;
    // stream next tile into the other buffer while this one is consumed
    if (kt + 1 < NKT) {
      int nxt = cur ^ 1;
      const float* gK = Kb + (long)((kt + 1) * 16 + skk) * HD + sd;
      const float* gV = Vb + (long)((kt + 1) * 16 + skk) * HD + sd;
      unsigned lK = (unsigned)(uintptr_t)&bufK[nxt][skk][sd];
      unsigned lV = (unsigned)(uintptr_t)&bufV[nxt][skk][sd];
      asm volatile("global_load_async_to_lds_b128 %0, %1, off"
                   :: "v"(lK), "v"(gK) : "memory");
      asm volatile("global_load_async_to_lds_b128 %0, %1, off"
                   :: "v"(lV), "v"(gV) : "memory");
      // async loads complete in order: <=2 outstanding => current tile landed
      asm volatile("s_wait_asynccnt 0x2" ::: "memory");
    } else {
      asm volatile("s_wait_asynccnt 0x0" ::: "memory");
    }
    __syncthreads();   // current tile visible to all 4 waves

    // S^T = K(16kk x 32d) * Q^T(32d x 16q) : 8 fp32 WMMAs, K from LDS
    v8f s = {0.f,0.f,0.f,0.f,0.f,0.f,0.f,0.f};
    const float* KrowL = &bufK[cur][ll][2 * hl];
    #pragma unroll
    for (int dc = 0; dc < 8; ++dc) {
      v2f aK = *(const v2f*)(KrowL + dc * 4);        // ds_load_b64
      s = wmma4(aK, qB[dc], s);
    }

    // online softmax over the key axis (8 in-lane values + half-swap)
    float P[8];
    float tm = -INFINITY;
    #pragma unroll
    for (int v = 0; v < 8; ++v) { P[v] = s[v] * invscale; tm = fmaxf(tm, P[v]); }
    tm = fmaxf(tm, __shfl_xor(tm, 16));
    float mnew  = fmaxf(mrun, tm);
    float alpha = __expf(mrun - mnew);
    float rsum = 0.f;
    #pragma unroll
    for (int v = 0; v < 8; ++v) { P[v] = __expf(P[v] - mnew); rsum += P[v]; }
    rsum += __shfl_xor(rsum, 16);
    lrun = lrun * alpha + rsum;
    mrun = mnew;
    #pragma unroll
    for (int v = 0; v < 8; ++v) { oLo[v] *= alpha; oHi[v] *= alpha; }

    // O^T += V^T * P^T : convert P (C layout) -> B layout with 1 shuffle+select
    #pragma unroll
    for (int kc = 0; kc < 4; ++kc) {
      int kb = kc * 4;
      v2f bp;
      if (kb < 8) {
        float s0 = __shfl_xor(P[kb + 2], 16);
        float s1 = __shfl_xor(P[kb + 3], 16);
        bp.x = (hl == 0) ? P[kb]     : s0;
        bp.y = (hl == 0) ? P[kb + 1] : s1;
      } else {
        float s0 = __shfl_xor(P[kb - 8], 16);
        float s1 = __shfl_xor(P[kb - 7], 16);
        bp.x = (hl == 0) ? s0 : P[kb - 6];
        bp.y = (hl == 0) ? s1 : P[kb - 5];
      }
      const float* vrowL = &bufV[cur][kb + 2 * hl][0];  // A = V^T : M=d, K=kk
      v2f aLo, aHi;
      aLo.x = vrowL[ll];       aLo.y = vrowL[HD + ll];       // kk, kk+1
      aHi.x = vrowL[16 + ll];  aHi.y = vrowL[HD + 16 + ll];
      oLo = wmma4(aLo, bp, oLo);
      oHi = wmma4(aHi, bp, oHi);
    }
    __syncthreads();   // all waves done reading before buffer is re-filled
  }

  // normalize and accumulate into fused[C][N]   (c = h*32 + d, column = q)
  float linv = 1.0f / lrun;
  float* fLo = fused + (long)(h * HD + 8 * hl) * NTOK + q;
  float* fHi = fused + (long)(h * HD + 16 + 8 * hl) * NTOK + q;
  #pragma unroll
  for (int v = 0; v < 8; ++v) {
    unsafeAtomicAdd(fLo + (long)v * NTOK, oLo[v] * linv);  // global_atomic_add_f32
    unsafeAtomicAdd(fHi + (long)v * NTOK, oHi[v] * linv);
  }
}

// ---------------------------------------------------------------------------
// Kernel 3: out projection + BatchNorm(eval) + ReLU, fused epilogue.
// ---------------------------------------------------------------------------
__global__ __launch_bounds__(256) void cma_outproj(
    const float* __restrict__ fused, const float* __restrict__ Wout,
    const float* __restrict__ gamma, const float* __restrict__ beta,
    const float* __restrict__ mean,  const float* __restrict__ var,
    float* __restrict__ out) {
  int wave = (blockIdx.x * blockDim.x + threadIdx.x) >> 5;
  int lane = threadIdx.x & 31;
  int nt = wave % NT16;
  int ot = wave / NT16;
  int hl = lane >> 4, ll = lane & 15;
  int n = nt * 16 + ll;
  int orow = ot * 16 + ll;

  v8f acc = {0.f, 0.f, 0.f, 0.f, 0.f, 0.f, 0.f, 0.f};
  for (int c0 = 0; c0 < CCH; c0 += 4) {
    int c = c0 + 2 * hl;
    v2f a = *(const v2f*)(Wout + orow * CCH + c);
    v2f b;
    b.x = fused[c * NTOK + n];
    b.y = fused[(c + 1) * NTOK + n];
    acc = wmma4(a, b, acc);
  }
  #pragma unroll
  for (int v = 0; v < 8; ++v) {
    int o = ot * 16 + 8 * hl + v;              // row of this C vgpr
    float inv = gamma[o] / sqrtf(var[o] + BNEPS);
    float y = acc[v] * inv + (beta[o] - mean[o] * inv);
    out[(long)o * NTOK + n] = fmaxf(y, 0.0f);
  }
}

// ---------------------------------------------------------------------------
extern "C" void kernel_launch(void* const* d_in, const int* in_sizes, int n_in,
                              void* d_out, int out_size, void* d_ws,
                              size_t ws_size, hipStream_t stream) {
  const float* f0    = (const float*)d_in[0];
  const float* f1    = (const float*)d_in[1];
  const float* f2    = (const float*)d_in[2];
  const float* f3    = (const float*)d_in[3];
  const float* Wq    = (const float*)d_in[4];
  const float* Wk    = (const float*)d_in[5];
  const float* Wv    = (const float*)d_in[6];
  const float* Wout  = (const float*)d_in[7];
  const float* gamma = (const float*)d_in[8];
  const float* beta  = (const float*)d_in[9];
  const float* mean  = (const float*)d_in[10];
  const float* var   = (const float*)d_in[11];

  // workspace layout (floats): Qt | Kt | Vt | fused  = 11.5 MB total
  const long PROJ = (long)MOD * NHD * NTOK * HD;  // 884736
  float* ws    = (float*)d_ws;
  float* Qt    = ws;
  float* Kt    = ws + PROJ;
  float* Vt    = ws + 2 * PROJ;
  float* fused = ws + 3 * PROJ;                   // 221184 floats
  const int FUSED_N = CCH * NTOK;

  cma_zero<<<(FUSED_N + 255) / 256, 256, 0, stream>>>(fused, FUSED_N);

  // 3 proj * 4 modalities * 8 o-tiles * 108 n-tiles = 10368 waves, 8/block
  cma_proj<<<10368 / 8, 256, 0, stream>>>(f0, f1, f2, f3, Wq, Wk, Wv,
                                          Qt, Kt, Vt);

  // 4 modalities * 4 heads * 27 blocks of 4 q-tile waves = 432 blocks
  cma_attn<<<(MOD * NHD) * (NT16 / 4), 128, 0, stream>>>(Qt, Kt, Vt, fused);

  // 8 o-tiles * 108 n-tiles = 864 waves, 8/block
  cma_outproj<<<864 / 8, 256, 0, stream>>>(fused, Wout, gamma, beta, mean, var,
                                           (float*)d_out);
}